// PALNET_topk_27118423507292
// MI455X (gfx1250) — compile-verified
//
#include <hip/hip_runtime.h>

#define TPB   256
#define BATCH 64
#define HROWS 50
#define WID0  500
#define ZROW  144000

typedef float v2f __attribute__((ext_vector_type(2)));
typedef float v8f __attribute__((ext_vector_type(8)));
typedef int   v4i __attribute__((ext_vector_type(4)));

#define AS1 __attribute__((address_space(1)))
#define AS3 __attribute__((address_space(3)))

#if defined(__gfx1250__) && __has_builtin(__builtin_amdgcn_global_load_async_to_lds_b128) && __has_builtin(__builtin_amdgcn_s_wait_asynccnt)
#define USE_ASYNC_LDS 1
#else
#define USE_ASYNC_LDS 0
#endif

__device__ __forceinline__ float reluf(float v) { return fmaxf(v, 0.0f); }

#if USE_ASYNC_LDS
// async copy of 16 bytes global -> LDS (ASYNCcnt-tracked)
__device__ __forceinline__ void async_copy16(const float* g, float* l)
{
  v4i* gp = (v4i*)(void*)g;
  v4i* lp = (v4i*)(void*)l;
  __builtin_amdgcn_global_load_async_to_lds_b128((AS1 v4i*)gp, (AS3 v4i*)lp, 0, 0);
}
#endif

// ---------------------------------------------------------------------------
// Generic 1x1-conv (+ReLU): input/weights in LDS, OB outputs per work item
// ---------------------------------------------------------------------------
template<int IC, int OC, int OB>
__device__ void conv_relu_lds(const float* sIn, float* sOut,
                              const float* sWgt, const float* sBia, int Wpix)
{
  const int OG = OC / OB;
  for (int it = threadIdx.x; it < Wpix * OG; it += TPB) {
    int p  = it / OG;
    int ob = (it - p * OG) * OB;
    float acc[OB];
    #pragma unroll
    for (int u = 0; u < OB; u++) acc[u] = sBia[ob + u];
    const float* ip = sIn + p * IC;
    #pragma unroll 4
    for (int i = 0; i < IC; i++) {
      float v = ip[i];
      const float* wr = sWgt + i * OC + ob;
      #pragma unroll
      for (int u = 0; u < OB; u++) acc[u] += v * wr[u];
    }
    float* op = sOut + p * OC + ob;
    #pragma unroll
    for (int u = 0; u < OB; u++) op[u] = reluf(acc[u]);
  }
}

// ---------------------------------------------------------------------------
// tanh-attention + top-10 gather (matches jax.lax.top_k: desc, low-idx ties)
// ---------------------------------------------------------------------------
template<int C>
__device__ void att_topk(const float* sX, const float* aW, const float* ab,
                         float* zout, int Wpix,
                         float* sE, float* sRv, int* sRi, int* sTI)
{
  for (int w = threadIdx.x; w < Wpix; w += TPB) {
    float a = ab[w];
    #pragma unroll 4
    for (int c = 0; c < C; c++) a += sX[w * C + c] * aW[c];   // aW idx uniform -> s_load
    sE[w] = tanhf(a);
  }
  __syncthreads();
  for (int it = 0; it < 10; it++) {
    float bv = -1e30f; int bi = 0x7fffffff;
    for (int w = threadIdx.x; w < Wpix; w += TPB) {
      float v = sE[w];
      if (v > bv || (v == bv && w < bi)) { bv = v; bi = w; }
    }
    sRv[threadIdx.x] = bv; sRi[threadIdx.x] = bi;
    __syncthreads();
    for (int s = TPB / 2; s > 0; s >>= 1) {
      if (threadIdx.x < s) {
        float ov = sRv[threadIdx.x + s]; int oi = sRi[threadIdx.x + s];
        if (ov > sRv[threadIdx.x] || (ov == sRv[threadIdx.x] && oi < sRi[threadIdx.x])) {
          sRv[threadIdx.x] = ov; sRi[threadIdx.x] = oi;
        }
      }
      __syncthreads();
    }
    if (threadIdx.x == 0) { sTI[it] = sRi[0]; sE[sRi[0]] = -1e30f; }
    __syncthreads();
  }
  for (int j = threadIdx.x; j < 10 * C; j += TPB) {
    int k = j / C, c = j - k * C;
    zout[j] = sX[sTI[k] * C + c];
  }
  __syncthreads();
}

// ---------------------------------------------------------------------------
// Backbone: one (b,h) row per workgroup; everything resident in LDS.
// Writes directly into concatenated z row: [xa1 | xa2 | xa3 | NCHW-flatten]
// ---------------------------------------------------------------------------
__global__ __launch_bounds__(TPB) void backbone_kernel(
    const float* __restrict__ x,
    const float* __restrict__ c1w1, const float* __restrict__ c1b1,
    const float* __restrict__ c1w2, const float* __restrict__ c1b2,
    const float* __restrict__ a1W,  const float* __restrict__ a1b,
    const float* __restrict__ c2w1, const float* __restrict__ c2b1,
    const float* __restrict__ c2w2, const float* __restrict__ c2b2,
    const float* __restrict__ a2W,  const float* __restrict__ a2b,
    const float* __restrict__ c3w1, const float* __restrict__ c3b1,
    const float* __restrict__ c3w2, const float* __restrict__ c3b2,
    const float* __restrict__ a3W,  const float* __restrict__ a3b,
    float* __restrict__ z)
{
  __shared__ float sX1[16384];  // x1(500x32) / x2(100x64) / conv3 weight stage
  __shared__ float sBuf[9600];  // xrow / p1 / t2 / t3+x3
  __shared__ float sP2[1280];   // p2 (20x64)
  __shared__ float sW[1280];    // small weight/bias stage
  __shared__ float sE[512];
  __shared__ float sRv[TPB];
  __shared__ int   sRi[TPB];
  __shared__ int   sTI[10];

  const int tid = threadIdx.x;
  const int b = blockIdx.x / HROWS;
  const int h = blockIdx.x % HROWS;
  float* zrow = z + (size_t)b * ZROW;

  // ---- load input row (500x3) + conv1 weights ----
  const float* xg = x + (size_t)(b * HROWS + h) * WID0 * 3;
  for (int i = tid; i < 1500; i += TPB) sBuf[i] = xg[i];
  for (int i = tid; i < 96;   i += TPB) sW[i]        = c1w1[i];
  for (int i = tid; i < 32;   i += TPB) sW[96 + i]   = c1b1[i];
  for (int i = tid; i < 1024; i += TPB) sW[128 + i]  = c1w2[i];
  for (int i = tid; i < 32;   i += TPB) sW[1152 + i] = c1b2[i];
  __syncthreads();

  // ---- conv1: 3->32 relu, 32->32 relu (fused per pixel) ----
  for (int p = tid; p < WID0; p += TPB) {
    float x0 = sBuf[p*3+0], x1 = sBuf[p*3+1], x2 = sBuf[p*3+2];
    float h1[32];
    #pragma unroll
    for (int o = 0; o < 32; o++)
      h1[o] = reluf(sW[96+o] + x0*sW[o] + x1*sW[32+o] + x2*sW[64+o]);
    #pragma unroll 4
    for (int o = 0; o < 32; o++) {
      float a = sW[1152+o];
      #pragma unroll
      for (int i = 0; i < 32; i++) a += h1[i] * sW[128 + i*32 + o];
      sX1[p*32 + o] = reluf(a);
    }
  }
  __syncthreads();

  att_topk<32>(sX1, a1W + h*32, a1b + h*WID0, zrow + h*320, WID0, sE, sRv, sRi, sTI);

  // ---- maxpool5: 500x32 -> 100x32 (sBuf) ----
  for (int j = tid; j < 100*32; j += TPB) {
    int w = j >> 5, c = j & 31;
    float m = -1e30f;
    #pragma unroll
    for (int p = 0; p < 5; p++) m = fmaxf(m, sX1[(w*5+p)*32 + c]);
    sBuf[j] = m;
  }
  __syncthreads();

  // ---- conv2: 32->64, 64->64 ----
  for (int i = tid; i < 2048; i += TPB) sX1[8192 + i] = c2w1[i];
  for (int i = tid; i < 64;   i += TPB) sW[i] = c2b1[i];
  __syncthreads();
  conv_relu_lds<32, 64, 8>(sBuf, sBuf + 3200, sX1 + 8192, sW, 100);
  __syncthreads();
  for (int i = tid; i < 4096; i += TPB) sX1[8192 + i] = c2w2[i];
  for (int i = tid; i < 64;   i += TPB) sW[64 + i] = c2b2[i];
  __syncthreads();
  conv_relu_lds<64, 64, 8>(sBuf + 3200, sX1, sX1 + 8192, sW + 64, 100);
  __syncthreads();

  att_topk<64>(sX1, a2W + h*64, a2b + h*100, zrow + 16000 + h*640, 100, sE, sRv, sRi, sTI);

  // ---- maxpool5: 100x64 -> 20x64 (sP2) ----
  for (int j = tid; j < 20*64; j += TPB) {
    int w = j >> 6, c = j & 63;
    float m = -1e30f;
    #pragma unroll
    for (int p = 0; p < 5; p++) m = fmaxf(m, sX1[(w*5+p)*64 + c]);
    sP2[j] = m;
  }
  __syncthreads();

  // ---- conv3: 64->128, 128->128 (x2 dead; reuse sX1 for weights) ----
  for (int i = tid; i < 8192; i += TPB) sX1[i] = c3w1[i];
  for (int i = tid; i < 128;  i += TPB) sW[i] = c3b1[i];
  __syncthreads();
  conv_relu_lds<64, 128, 8>(sP2, sBuf, sX1, sW, 20);
  __syncthreads();
  for (int i = tid; i < 16384; i += TPB) sX1[i] = c3w2[i];
  for (int i = tid; i < 128;   i += TPB) sW[128 + i] = c3b2[i];
  __syncthreads();
  conv_relu_lds<128, 128, 8>(sBuf, sBuf + 2560, sX1, sW + 128, 20);
  __syncthreads();

  att_topk<128>(sBuf + 2560, a3W + h*128, a3b + h*20, zrow + 48000 + h*1280, 20, sE, sRv, sRi, sTI);

  // ---- final maxpool4: 20x128 -> 5x128, NCHW-transposed flatten ----
  for (int j = tid; j < 640; j += TPB) {
    int c = j / 5, w = j - 5 * c;
    float m = -1e30f;
    #pragma unroll
    for (int p = 0; p < 4; p++) m = fmaxf(m, sBuf[2560 + (w*4+p)*128 + c]);
    zrow[112000 + c*250 + h*5 + w] = m;
  }
}

// ---------------------------------------------------------------------------
// f32 WMMA GEMM: C(64x1024) = A(64xK) * W(Kx1024), split-K capable.
// KSTEP=32, double-buffered LDS, async global->LDS copies when available.
// 8 waves/block: 4 M-tiles x 2 N-subtile groups; V_WMMA_F32_16X16X4_F32.
// ---------------------------------------------------------------------------
#define KSTEP 32
#define LA_STRIDE 36   // 32 + pad: float4-aligned, conflict-free frag reads
#define LB_STRIDE 72   // 64 + pad: halves offset by 16 banks per 2 K-rows

__device__ __forceinline__ void wmma_compute32(const float* lAb, const float* lBb,
                                               int m0, int nt, int half, int lm,
                                               v8f& acc0, v8f& acc1)
{
  #pragma unroll
  for (int j = 0; j < 8; j++) {
    const int ka = j * 4 + half * 2;   // lanes 0-15: K0,K1; lanes 16-31: K2,K3
    v2f a  = { lAb[(m0 + lm) * LA_STRIDE + ka], lAb[(m0 + lm) * LA_STRIDE + ka + 1] };
    v2f b0 = { lBb[ka * LB_STRIDE + nt + lm],      lBb[(ka + 1) * LB_STRIDE + nt + lm] };
    v2f b1 = { lBb[ka * LB_STRIDE + nt + 16 + lm], lBb[(ka + 1) * LB_STRIDE + nt + 16 + lm] };
    acc0 = __builtin_amdgcn_wmma_f32_16x16x4_f32(false, a, false, b0, (short)0, acc0, false, false);
    acc1 = __builtin_amdgcn_wmma_f32_16x16x4_f32(false, a, false, b1, (short)0, acc1, false, false);
  }
}

__global__ __launch_bounds__(TPB) void wmma_fc_kernel(
    const float* __restrict__ A, const float* __restrict__ Wt,
    float* __restrict__ Out, const float* __restrict__ bias,
    int Kstride, int kLen, int outSplitStride, int applyRelu)
{
  __shared__ float lA[2][64 * LA_STRIDE];
  __shared__ float lB[2][KSTEP * LB_STRIDE];

  const int tid = threadIdx.x, lane = tid & 31, wave = tid >> 5;
  const int n0 = blockIdx.x * 64;
  const int k0 = blockIdx.y * kLen;
  Out += (size_t)blockIdx.y * outSplitStride;
  const int m0   = (wave & 3) * 16;
  const int nt   = (wave >> 2) * 32;
  const int half = lane >> 4, lm = lane & 15;

  v8f acc0 = {0.f,0.f,0.f,0.f,0.f,0.f,0.f,0.f};
  v8f acc1 = {0.f,0.f,0.f,0.f,0.f,0.f,0.f,0.f};

  // per-thread cooperative-load coordinates (2 float4 each for A and B tiles)
  const int ra0 = tid >> 3,          ca0 = (tid & 7) * 4;          // A item 0
  const int ra1 = (tid + TPB) >> 3,  ca1 = ((tid + TPB) & 7) * 4;  // A item 1
  const int rb0 = tid >> 4,          cb0 = (tid & 15) * 4;         // B item 0
  const int rb1 = (tid + TPB) >> 4,  cb1 = ((tid + TPB) & 15) * 4; // B item 1

#if USE_ASYNC_LDS
  // ---- async global->LDS pipeline (ASYNCcnt), single barrier per step ----
  int buf = 0;
  {
    async_copy16(A  + (size_t)ra0 * Kstride + k0 + ca0,       &lA[0][ra0 * LA_STRIDE + ca0]);
    async_copy16(A  + (size_t)ra1 * Kstride + k0 + ca1,       &lA[0][ra1 * LA_STRIDE + ca1]);
    async_copy16(Wt + (size_t)(k0 + rb0) * 1024 + n0 + cb0,   &lB[0][rb0 * LB_STRIDE + cb0]);
    async_copy16(Wt + (size_t)(k0 + rb1) * 1024 + n0 + cb1,   &lB[0][rb1 * LB_STRIDE + cb1]);
  }
  for (int kc = 0; kc < kLen; kc += KSTEP) {
    __builtin_amdgcn_s_wait_asynccnt(0);   // tile `buf` fully landed in LDS
    __syncthreads();                       // all waves past compute of buf^1
    if (kc + KSTEP < kLen) {               // issue next tile; overlaps compute
      const int k = k0 + kc + KSTEP;
      const int nb = buf ^ 1;
      async_copy16(A  + (size_t)ra0 * Kstride + k + ca0,      &lA[nb][ra0 * LA_STRIDE + ca0]);
      async_copy16(A  + (size_t)ra1 * Kstride + k + ca1,      &lA[nb][ra1 * LA_STRIDE + ca1]);
      async_copy16(Wt + (size_t)(k + rb0) * 1024 + n0 + cb0,  &lB[nb][rb0 * LB_STRIDE + cb0]);
      async_copy16(Wt + (size_t)(k + rb1) * 1024 + n0 + cb1,  &lB[nb][rb1 * LB_STRIDE + cb1]);
    }
    wmma_compute32(lA[buf], lB[buf], m0, nt, half, lm, acc0, acc1);
    buf ^= 1;
  }
#else
  // ---- fallback: register-prefetch software pipeline ----
  float4 va0, va1, vb0, vb1;
  {
    va0 = *(const float4*)(A + (size_t)ra0 * Kstride + k0 + ca0);
    va1 = *(const float4*)(A + (size_t)ra1 * Kstride + k0 + ca1);
    vb0 = *(const float4*)(Wt + (size_t)(k0 + rb0) * 1024 + n0 + cb0);
    vb1 = *(const float4*)(Wt + (size_t)(k0 + rb1) * 1024 + n0 + cb1);
  }
  int buf = 0;
  for (int kc = 0; kc < kLen; kc += KSTEP) {
    *(float4*)&lA[buf][ra0 * LA_STRIDE + ca0] = va0;
    *(float4*)&lA[buf][ra1 * LA_STRIDE + ca1] = va1;
    *(float4*)&lB[buf][rb0 * LB_STRIDE + cb0] = vb0;
    *(float4*)&lB[buf][rb1 * LB_STRIDE + cb1] = vb1;
    __syncthreads();
    if (kc + KSTEP < kLen) {   // issue next-tile loads; consumed next iteration
      const int k = k0 + kc + KSTEP;
      va0 = *(const float4*)(A + (size_t)ra0 * Kstride + k + ca0);
      va1 = *(const float4*)(A + (size_t)ra1 * Kstride + k + ca1);
      vb0 = *(const float4*)(Wt + (size_t)(k + rb0) * 1024 + n0 + cb0);
      vb1 = *(const float4*)(Wt + (size_t)(k + rb1) * 1024 + n0 + cb1);
    }
    wmma_compute32(lA[buf], lB[buf], m0, nt, half, lm, acc0, acc1);
    buf ^= 1;
  }
#endif

  #pragma unroll
  for (int r = 0; r < 8; r++) {     // C layout: VGPR r -> M=r (+8 for upper half)
    int m = m0 + r + half * 8;
    int n = n0 + nt + lm;
    float v0 = acc0[r], v1 = acc1[r];
    if (bias) { v0 += bias[n]; v1 += bias[n + 16]; }
    if (applyRelu) { v0 = fmaxf(v0, 0.f); v1 = fmaxf(v1, 0.f); }
    Out[(size_t)m * 1024 + n]      = v0;
    Out[(size_t)m * 1024 + n + 16] = v1;
  }
}

// ---------------------------------------------------------------------------
// Deterministic split-K reduction for fc1 (+bias, +ReLU)
// ---------------------------------------------------------------------------
__global__ __launch_bounds__(TPB) void reduce_fc1_kernel(
    const float* __restrict__ P, const float* __restrict__ bias, float* __restrict__ z1)
{
  int idx = blockIdx.x * TPB + threadIdx.x;   // 0..65535
  float s = 0.f;
  for (int t = 0; t < 125; t++) s += P[(size_t)t * 65536 + idx];
  z1[idx] = fmaxf(s + bias[idx & 1023], 0.f);
}

// ---------------------------------------------------------------------------
// fc3: (64x1024) @ (1024x150) + bias
// ---------------------------------------------------------------------------
__global__ __launch_bounds__(192) void fc3_kernel(
    const float* __restrict__ z2, const float* __restrict__ w,
    const float* __restrict__ bias, float* __restrict__ out)
{
  __shared__ float sZ[1024];
  const int b = blockIdx.x;
  for (int i = threadIdx.x; i < 1024; i += 192) sZ[i] = z2[b * 1024 + i];
  __syncthreads();
  const int n = threadIdx.x;
  if (n < 150) {
    float acc = bias[n];
    for (int k = 0; k < 1024; k++) acc += sZ[k] * w[k * 150 + n];
    out[b * 150 + n] = acc;
  }
}

// ---------------------------------------------------------------------------
extern "C" void kernel_launch(void* const* d_in, const int* in_sizes, int n_in,
                              void* d_out, int out_size, void* d_ws, size_t ws_size,
                              hipStream_t stream)
{
  (void)in_sizes; (void)n_in; (void)out_size; (void)ws_size;
  const float* x    = (const float*)d_in[0];
  const float* c1w1 = (const float*)d_in[1];
  const float* c1b1 = (const float*)d_in[2];
  const float* c1w2 = (const float*)d_in[3];
  const float* c1b2 = (const float*)d_in[4];
  const float* a1W  = (const float*)d_in[5];
  const float* a1b  = (const float*)d_in[6];
  const float* c2w1 = (const float*)d_in[7];
  const float* c2b1 = (const float*)d_in[8];
  const float* c2w2 = (const float*)d_in[9];
  const float* c2b2 = (const float*)d_in[10];
  const float* a2W  = (const float*)d_in[11];
  const float* a2b  = (const float*)d_in[12];
  const float* c3w1 = (const float*)d_in[13];
  const float* c3b1 = (const float*)d_in[14];
  const float* c3w2 = (const float*)d_in[15];
  const float* c3b2 = (const float*)d_in[16];
  const float* a3W  = (const float*)d_in[17];
  const float* a3b  = (const float*)d_in[18];
  const float* fc1w = (const float*)d_in[19];
  const float* fc1b = (const float*)d_in[20];
  const float* fc2w = (const float*)d_in[21];
  const float* fc2b = (const float*)d_in[22];
  const float* fc3w = (const float*)d_in[23];
  const float* fc3b = (const float*)d_in[24];

  float* ws = (float*)d_ws;
  float* z  = ws;                               // 64 x 144000
  float* P  = ws + (size_t)9216000;             // 125 x 64 x 1024 partials
  float* z1 = ws + (size_t)9216000 + 8192000;   // 64 x 1024
  float* z2 = z1 + 65536;                       // 64 x 1024

  backbone_kernel<<<dim3(BATCH * HROWS), dim3(TPB), 0, stream>>>(
      x, c1w1, c1b1, c1w2, c1b2, a1W, a1b,
      c2w1, c2b1, c2w2, c2b2, a2W, a2b,
      c3w1, c3b1, c3w2, c3b2, a3W, a3b, z);

  // fc1: split-K (125 chunks of 1152) -> partials, then deterministic reduce+bias+ReLU
  wmma_fc_kernel<<<dim3(16, 125), dim3(TPB), 0, stream>>>(
      z, fc1w, P, nullptr, 144000, 1152, 65536, 0);
  reduce_fc1_kernel<<<dim3(256), dim3(TPB), 0, stream>>>(P, fc1b, z1);

  // fc2: single pass, +bias, no activation
  wmma_fc_kernel<<<dim3(16, 1), dim3(TPB), 0, stream>>>(
      z1, fc2w, z2, fc2b, 1024, 1024, 0, 0);

  fc3_kernel<<<dim3(64), dim3(192), 0, stream>>>(z2, fc3w, fc3b, (float*)d_out);
}